// XMetaPath_54614804136170
// MI455X (gfx1250) — compile-verified
//
#include <hip/hip_runtime.h>
#include <hip/hip_bf16.h>

#define HDIM 128
#define NNODES 50000
#define NEDGES 640000
#define TILE_M 16

typedef __attribute__((ext_vector_type(2))) float v2f;
typedef __attribute__((ext_vector_type(8))) float v8f;

// ---------------------------------------------------------------------------
// K-loop of V_WMMA_F32_16X16X4_F32: D += A(16x4) * B(4x16) over K=128.
// Arow = feature row base for this lane (A + node*H), scaled by ascale.
// Wrow = weight row base for this lane's output column (W + col*H); since
// B[k][n] = W[n][k], the B fragment is a contiguous float2 from W's row n.
// off2 = 2*(lane>>4) selects the K pair this lane owns for both A and B.
// ---------------------------------------------------------------------------
__device__ __forceinline__ v8f wmma_accum(const float* __restrict__ Arow,
                                          float ascale,
                                          const float* __restrict__ Wrow,
                                          int off2, v8f acc) {
#pragma unroll
    for (int kk = 0; kk < HDIM; kk += 4) {
        v2f a = *(const v2f*)(Arow + kk + off2);
        a.x *= ascale;
        a.y *= ascale;
        v2f b = *(const v2f*)(Wrow + kk + off2);
        acc = __builtin_amdgcn_wmma_f32_16x16x4_f32(
            /*neg_a=*/false, a, /*neg_b=*/false, b,
            /*c_mod=*/(short)0, acc, /*reuse_a=*/false, /*reuse_b=*/false);
    }
    return acc;
}

// ---------------------------------------------------------------------------
// Prep: fold gate into weights/bias, compute lambda = softplus(raw_lam)+1e-8.
// ---------------------------------------------------------------------------
__global__ void mp_prep_kernel(const float* __restrict__ bl,
                               const float* __restrict__ W0, const float* __restrict__ b0,
                               const float* __restrict__ W1, const float* __restrict__ b1,
                               const float* __restrict__ gate, const float* __restrict__ raw_lam,
                               float* __restrict__ W0s, float* __restrict__ W1s,
                               float* __restrict__ biasC, float* __restrict__ lam) {
    int i = blockIdx.x * blockDim.x + threadIdx.x;
    const int HH = HDIM * HDIM;
    if (i >= 3 * HH) return;
    int layer = i / HH;
    int idx = i - layer * HH;
    float g = 1.0f / (1.0f + expf(-gate[layer]));
    W0s[i] = (1.0f - g) * W0[i];
    W1s[i] = g * W1[i];
    if (idx < HDIM) {
        biasC[layer * HDIM + idx] = bl[layer * HDIM + idx]
                                  + (1.0f - g) * b0[layer * HDIM + idx]
                                  + g * b1[layer * HDIM + idx];
    }
    if (idx == 0) {
        float x = raw_lam[layer];
        float sp = (x > 20.0f) ? x : log1pf(expf(x));
        lam[layer] = sp + 1e-8f;
    }
}

__global__ void mp_zero_kernel(float* __restrict__ p, size_t n) {
    size_t i = (size_t)blockIdx.x * blockDim.x + threadIdx.x;
    if (i < n) p[i] = 0.0f;
}

// ---------------------------------------------------------------------------
// Edge scatter: one wave32 per edge. Lane l owns channels 4l..4l+3 (float4
// gather from src row, 4 f32 global atomic adds to dst row). Lane 0 adds w
// to the degree accumulator.
// ---------------------------------------------------------------------------
__global__ void __launch_bounds__(256)
mp_scatter_kernel(const float* __restrict__ hsrc,
                  const float* __restrict__ tsrc, const float* __restrict__ tdst,
                  const int* __restrict__ es, const int* __restrict__ ed,
                  const float* __restrict__ lam_p,
                  float* __restrict__ agg, float* __restrict__ deg, int nedges) {
    int tid = blockIdx.x * blockDim.x + threadIdx.x;
    int e = tid >> 5;
    int lane = tid & 31;
    if (e >= nedges) return;
    int s = es[e];
    int d = ed[e];
    float delta = fmaxf(tdst[d] - tsrc[s], 0.0f);
    float lam = lam_p[0];
    float w = expf(fmaxf(-lam * delta, -60.0f));
    const float4 hv = *(const float4*)(hsrc + (size_t)s * HDIM + lane * 4);
    float* ap = agg + (size_t)d * HDIM + lane * 4;
    atomicAdd(ap + 0, w * hv.x);
    atomicAdd(ap + 1, w * hv.y);
    atomicAdd(ap + 2, w * hv.z);
    atomicAdd(ap + 3, w * hv.w);
    if (lane == 0) atomicAdd(deg + d, w);
}

// ---------------------------------------------------------------------------
// Node update: 16-node tile per block, 8 waves x 16-column WMMA tiles.
// h = (agg/deg)@Wl^T + hdst@W0s^T + x0@W1s^T + biasC ; relu ; layernorm ;
// gated write (deg>0 => updated, else passthrough hdst).
// ---------------------------------------------------------------------------
__global__ void __launch_bounds__(256)
mp_update_kernel(const float* __restrict__ agg, const float* __restrict__ deg,
                 const float* __restrict__ hdst, const float* __restrict__ x0,
                 const float* __restrict__ Wl, const float* __restrict__ W0s,
                 const float* __restrict__ W1s, const float* __restrict__ biasC,
                 const float* __restrict__ lng, const float* __restrict__ lnb,
                 float* __restrict__ hout) {
    __shared__ float smem[TILE_M][HDIM + 4];
    __shared__ float psum[TILE_M][16];
    __shared__ float psq[TILE_M][16];
    __shared__ float s_mean[TILE_M];
    __shared__ float s_inv[TILE_M];

    const int tid = threadIdx.x;
    const int wave = tid >> 5;         // 0..7 -> 16-wide column tile
    const int lane = tid & 31;
    const int m = lane & 15;           // A row / B column within tile
    const int off2 = (lane >> 4) * 2;  // K sub-offset owned by this lane
    const int node = blockIdx.x * TILE_M + m;
    const int col = wave * 16 + m;     // global output channel

    const float rdeg = 1.0f / fmaxf(deg[node], 1e-6f);

    v8f acc = {};
    acc = wmma_accum(agg  + (size_t)node * HDIM, rdeg, Wl  + (size_t)col * HDIM, off2, acc);
    acc = wmma_accum(hdst + (size_t)node * HDIM, 1.0f, W0s + (size_t)col * HDIM, off2, acc);
    acc = wmma_accum(x0   + (size_t)node * HDIM, 1.0f, W1s + (size_t)col * HDIM, off2, acc);

    const float bc = biasC[col];
#pragma unroll
    for (int j = 0; j < 8; ++j) {
        int row = j + (lane >> 4) * 8;   // D layout: VGPR j -> row j (+8 for hi lanes)
        smem[row][col] = fmaxf(acc[j] + bc, 0.0f);
    }
    __syncthreads();

    // LayerNorm stats: 16 threads per row, 8 channels each.
    {
        int r = tid >> 4;
        int seg = tid & 15;
        const float* p = &smem[r][seg * 8];
        float s = 0.0f, q = 0.0f;
#pragma unroll
        for (int i = 0; i < 8; ++i) { float v = p[i]; s += v; q += v * v; }
        psum[r][seg] = s;
        psq[r][seg] = q;
    }
    __syncthreads();
    if (tid < TILE_M) {
        float s = 0.0f, q = 0.0f;
#pragma unroll
        for (int i = 0; i < 16; ++i) { s += psum[tid][i]; q += psq[tid][i]; }
        float mean = s * (1.0f / HDIM);
        float var = q * (1.0f / HDIM) - mean * mean;
        s_mean[tid] = mean;
        s_inv[tid] = rsqrtf(fmaxf(var, 0.0f) + 1e-5f);
    }
    __syncthreads();
    {
        int r = tid >> 4;
        int seg = tid & 15;
        int nd = blockIdx.x * TILE_M + r;
        float mean = s_mean[r];
        float inv = s_inv[r];
        bool upd = deg[nd] > 0.0f;
#pragma unroll
        for (int i = 0; i < 8; ++i) {
            int c = seg * 8 + i;
            float y = upd ? (smem[r][c] - mean) * inv * lng[c] + lnb[c]
                          : hdst[(size_t)nd * HDIM + c];
            hout[(size_t)nd * HDIM + c] = y;
        }
    }
}

// ---------------------------------------------------------------------------
// Final projection: d_out = hB @ out_W^T + out_b (single WMMA GEMM).
// ---------------------------------------------------------------------------
__global__ void __launch_bounds__(256)
mp_outproj_kernel(const float* __restrict__ hin, const float* __restrict__ W,
                  const float* __restrict__ b, float* __restrict__ out) {
    const int tid = threadIdx.x;
    const int wave = tid >> 5;
    const int lane = tid & 31;
    const int m = lane & 15;
    const int off2 = (lane >> 4) * 2;
    const int node = blockIdx.x * TILE_M + m;
    const int col = wave * 16 + m;

    v8f acc = {};
    acc = wmma_accum(hin + (size_t)node * HDIM, 1.0f, W + (size_t)col * HDIM, off2, acc);
    const float bc = b[col];
#pragma unroll
    for (int j = 0; j < 8; ++j) {
        int row = j + (lane >> 4) * 8;
        int nd = blockIdx.x * TILE_M + row;
        out[(size_t)nd * HDIM + col] = acc[j] + bc;
    }
}

extern "C" void kernel_launch(void* const* d_in, const int* in_sizes, int n_in,
                              void* d_out, int out_size, void* d_ws, size_t ws_size,
                              hipStream_t stream) {
    const float* x_A   = (const float*)d_in[0];
    const float* x_B   = (const float*)d_in[1];
    const float* t_A   = (const float*)d_in[2];
    const float* t_B   = (const float*)d_in[3];
    const float* Wl    = (const float*)d_in[4];
    const float* bl    = (const float*)d_in[5];
    const float* W0    = (const float*)d_in[6];
    const float* b0    = (const float*)d_in[7];
    const float* W1    = (const float*)d_in[8];
    const float* b1    = (const float*)d_in[9];
    const float* gate  = (const float*)d_in[10];
    const float* rlam  = (const float*)d_in[11];
    const float* ln_g  = (const float*)d_in[12];
    const float* ln_b  = (const float*)d_in[13];
    const float* out_W = (const float*)d_in[14];
    const float* out_b = (const float*)d_in[15];
    const int* e_src[3] = {(const int*)d_in[16], (const int*)d_in[18], (const int*)d_in[20]};
    const int* e_dst[3] = {(const int*)d_in[17], (const int*)d_in[19], (const int*)d_in[21]};

    const size_t NH = (size_t)NNODES * HDIM;
    const int HH = HDIM * HDIM;
    float* ws    = (float*)d_ws;
    float* hB    = ws;                 // N*H
    float* hA    = ws + NH;            // N*H
    float* agg   = ws + 2 * NH;        // N*H   (contiguous with deg for zeroing)
    float* deg   = ws + 3 * NH;        // N
    float* W0s   = deg + NNODES;       // 3*H*H
    float* W1s   = W0s + 3 * HH;       // 3*H*H
    float* biasC = W1s + 3 * HH;       // 3*H
    float* lam   = biasC + 3 * HDIM;   // 3

    mp_prep_kernel<<<(3 * HH + 255) / 256, 256, 0, stream>>>(
        bl, W0, b0, W1, b1, gate, rlam, W0s, W1s, biasC, lam);

    struct Layer {
        const float* hsrc; const float* ts; const float* td;
        int li; const float* hdst; const float* x0; float* hout;
    };
    // reference processes metapath in reversed order: r2, r1, r0
    Layer layers[3] = {
        { x_A, t_A, t_B, 2, x_B, x_B, hB },
        { hB,  t_B, t_A, 1, x_A, x_A, hA },
        { hA,  t_A, t_B, 0, hB,  x_B, hB },   // in-place on hB (barrier-safe)
    };

    const size_t zn = NH + NNODES;
    const int zgrid = (int)((zn + 255) / 256);
    const int sgrid = (int)(((size_t)NEDGES * 32 + 255) / 256);
    const int ngrid = NNODES / TILE_M;

    for (int i = 0; i < 3; ++i) {
        const Layer& L = layers[i];
        mp_zero_kernel<<<zgrid, 256, 0, stream>>>(agg, zn);
        mp_scatter_kernel<<<sgrid, 256, 0, stream>>>(
            L.hsrc, L.ts, L.td, e_src[L.li], e_dst[L.li], lam + L.li, agg, deg, NEDGES);
        mp_update_kernel<<<ngrid, 256, 0, stream>>>(
            agg, deg, L.hdst, L.x0,
            Wl + (size_t)L.li * HH, W0s + (size_t)L.li * HH, W1s + (size_t)L.li * HH,
            biasC + (size_t)L.li * HDIM, ln_g + (size_t)L.li * HDIM, ln_b + (size_t)L.li * HDIM,
            L.hout);
    }

    mp_outproj_kernel<<<ngrid, 256, 0, stream>>>(hB, out_W, out_b, (float*)d_out);
}